// MaxUnpooling2D_661424963976
// MI455X (gfx1250) — compile-verified
//
#include <hip/hip_runtime.h>

// MaxUnpooling2D for MI455X (gfx1250): memory-bound scatter-add.
// B=8, H=128, W=128, C=256, pool 2x2 -> out (8,256,256,256) f32.
// Per-batch input block  = 128*128*256 = 2^22 elements.
// Per-batch output block = 256*256*256 = 2^24 elements.

typedef float v4f __attribute__((ext_vector_type(4)));
typedef int   v4i __attribute__((ext_vector_type(4)));

// ---------------------------------------------------------------------------
// Kernel 1: zero the 512 MB output with 128-bit stores (regular temporal hint:
// we WANT these lines resident in the 192 MB L2 when the atomics arrive).
// ---------------------------------------------------------------------------
__global__ __launch_bounds__(256) void unpool_zero_b128(v4f* __restrict__ out, int n4) {
    int t = blockIdx.x * blockDim.x + threadIdx.x;
    if (t < n4) {
        v4f z = {0.0f, 0.0f, 0.0f, 0.0f};
        out[t] = z;
    }
}

// ---------------------------------------------------------------------------
// Device-scope, non-returning hardware FP32 atomic add.
// SADDR form: 64-bit uniform base in SGPRs + 32-bit per-lane byte offset.
// scope:SCOPE_DEV is required for cross-WGP coherence of the RMW;
// default TH (RT, non-returning) keeps it on STOREcnt only.
// ---------------------------------------------------------------------------
__device__ __forceinline__ void atomic_add_f32_dev(float* base, unsigned byte_off, float val) {
    asm volatile("global_atomic_add_f32 %0, %1, %2 scope:SCOPE_DEV"
                 :
                 : "v"(byte_off), "v"(val), "s"(base)
                 : "memory");
}

// ---------------------------------------------------------------------------
// Kernel 2: each thread handles 4 consecutive elements.
//  - float4 / int4 loads, non-temporal (read-once 256 MB stream must not
//    evict output lines from L2, where the atomics execute).
//  - 4 device-scope fp32 atomics into the output.
// All 4 elements of a float4 share one batch: 2^20 float4's per batch.
// ---------------------------------------------------------------------------
__global__ __launch_bounds__(256) void unpool_scatter_add(const v4f* __restrict__ in,
                                                          const v4i* __restrict__ idx,
                                                          float* __restrict__ out,
                                                          int n4) {
    int t = blockIdx.x * blockDim.x + threadIdx.x;
    if (t >= n4) return;

    v4f v  = __builtin_nontemporal_load(in  + t);   // global_load_b128 th:TH_LOAD_NT
    v4i ix = __builtin_nontemporal_load(idx + t);   // global_load_b128 th:TH_LOAD_NT

    // batch = (element_index >> 22) = (t*4) >> 22 = t >> 20
    unsigned boff_elems = (((unsigned)t) >> 20) << 24;  // batch * 2^24

    atomic_add_f32_dev(out, ((unsigned)ix.x + boff_elems) * 4u, v.x);
    atomic_add_f32_dev(out, ((unsigned)ix.y + boff_elems) * 4u, v.y);
    atomic_add_f32_dev(out, ((unsigned)ix.z + boff_elems) * 4u, v.z);
    atomic_add_f32_dev(out, ((unsigned)ix.w + boff_elems) * 4u, v.w);
}

extern "C" void kernel_launch(void* const* d_in, const int* in_sizes, int n_in,
                              void* d_out, int out_size, void* d_ws, size_t ws_size,
                              hipStream_t stream) {
    const float* in  = (const float*)d_in[0];   // (8,128,128,256) f32
    const int*   idx = (const int*)  d_in[1];   // (8,128,128,256) i32
    float*       out = (float*)d_out;           // (8,256,256,256) f32

    const int n  = in_sizes[0];     // 33,554,432 elements (divisible by 4)
    const int n4 = n >> 2;          // 8,388,608 float4's
    const int o4 = out_size >> 2;   // 33,554,432 float4's (512 MB)

    // 1) zero the output (every call: harness poisons d_out and replays)
    {
        dim3 blk(256);
        dim3 grd((o4 + 255) / 256);
        unpool_zero_b128<<<grd, blk, 0, stream>>>((v4f*)out, o4);
    }
    // 2) scatter-add (stream-ordered after the zero fill)
    {
        dim3 blk(256);
        dim3 grd((n4 + 255) / 256);
        unpool_scatter_add<<<grd, blk, 0, stream>>>((const v4f*)in, (const v4i*)idx, out, n4);
    }
}